// FFTConvNet_27805618274529
// MI455X (gfx1250) — compile-verified
//
#include <hip/hip_runtime.h>
#include <cstddef>

typedef __attribute__((ext_vector_type(2))) float vf2;
typedef __attribute__((ext_vector_type(8))) float vf8;

#define PI2 6.283185307179586f

// Problem constants
#define BB   4
#define CIN  64
#define COUT 64
#define HH   128
#define WW   128
#define WF   65              // W/2+1
#define FREQ (HH * WF)       // 8320
#define MROWS (BB * FREQ)    // 33280

// ---------------------------------------------------------------------------
// 128-point iterative radix-2 FFT in LDS. blockDim.x == 64, 2 points/thread.
// sign = -1 forward, +1 inverse (unscaled).
// ---------------------------------------------------------------------------
__device__ __forceinline__ void fft128_shared(float2* s, int t, float sign) {
    __syncthreads();
    // bit-reversal permutation (7-bit)
#pragma unroll
    for (int i = 0; i < 2; ++i) {
        int idx = t + (i << 6);
        int j = (int)(__brev((unsigned)idx) >> 25);
        if (j > idx) { float2 tmp = s[idx]; s[idx] = s[j]; s[j] = tmp; }
    }
    __syncthreads();
#pragma unroll
    for (int st = 1; st <= 7; ++st) {
        int half = 1 << (st - 1);
        int grp  = t >> (st - 1);
        int idx  = t & (half - 1);
        int p    = (grp << st) + idx;
        float ang = sign * PI2 * (float)idx / (float)(1 << st);
        float wr, wi;
        __sincosf(ang, &wi, &wr);
        float2 u = s[p], v = s[p + half];
        float tr = v.x * wr - v.y * wi;
        float ti = v.x * wi + v.y * wr;
        s[p].x        = u.x + tr; s[p].y        = u.y + ti;
        s[p + half].x = u.x - tr; s[p + half].y = u.y - ti;
        __syncthreads();
    }
}

// ---------------------------------------------------------------------------
// Stage 1: row rFFT.  one block per (b*Cin*H + row).  Xrow[(bc*128+row)*65+fx]
// ---------------------------------------------------------------------------
__global__ void __launch_bounds__(64) k_fft_rows(const float* __restrict__ x,
                                                 float2* __restrict__ Xrow) {
    __shared__ float2 s[128];
    int blk = blockIdx.x;
    int t = threadIdx.x;
    const float* xr = x + (size_t)blk * WW;
    s[t]      = make_float2(xr[t], 0.0f);
    s[t + 64] = make_float2(xr[t + 64], 0.0f);
    fft128_shared(s, t, -1.0f);
    float2* out = Xrow + (size_t)blk * WF;
    out[t] = s[t];
    if (t == 0) out[64] = s[64];
}

// ---------------------------------------------------------------------------
// Stage 2: column FFT over H, then fftshift + high-pass mask, scatter into
// A_re/A_im laid out [M = b*FREQ + fys*65 + fxs][K = cin]  (row-major, K=64)
// ---------------------------------------------------------------------------
__global__ void __launch_bounds__(64) k_fft_cols(const float2* __restrict__ Xrow,
                                                 float* __restrict__ Are,
                                                 float* __restrict__ Aim) {
    __shared__ float2 s[128];
    int blk = blockIdx.x;            // bc*65 + fx
    int bc = blk / WF, fx = blk % WF;
    int b = bc >> 6, c = bc & 63;
    int t = threadIdx.x;
    const float2* col = Xrow + (size_t)bc * FREQ + fx;
    s[t]      = col[(size_t)t * WF];
    s[t + 64] = col[(size_t)(t + 64) * WF];
    fft128_shared(s, t, -1.0f);
    int fxs = fx + 33; if (fxs >= WF) fxs -= WF;     // fftshift along rfft axis
    float dx = (float)(fxs - 64);
#pragma unroll
    for (int i = 0; i < 2; ++i) {
        int fy = t + (i << 6);
        int fys = (fy + 64) & 127;                   // fftshift along H axis
        float dy = (float)(fys - 64);
        bool keep = (dx * dx + dy * dy) > 900.0f;    // high-pass r > 30
        float2 v = s[fy];
        if (!keep) v = make_float2(0.0f, 0.0f);
        size_t rowM = (size_t)b * FREQ + (size_t)fys * WF + fxs;
        Are[rowM * 64 + c] = v.x;
        Aim[rowM * 64 + c] = v.y;
    }
}

// ---------------------------------------------------------------------------
// Stage 3: weight transpose:  Wt[tap][n=o][k=i] = w[o][i][tap]
// ---------------------------------------------------------------------------
__global__ void k_prep_w(const float* __restrict__ w, float* __restrict__ Wt) {
    int idx = blockIdx.x * blockDim.x + threadIdx.x;
    if (idx >= 9 * 64 * 64) return;
    int tap = idx >> 12;
    int rem = idx & 4095;
    int n = rem >> 6, k = rem & 63;
    Wt[idx] = w[(size_t)(n * 64 + k) * 9 + tap];
}

// ---------------------------------------------------------------------------
// Async DMA of one 16KB W_tap matrix into an LDS buffer.
// Uses CDNA5 GLOBAL_LOAD_ASYNC_TO_LDS_B128 (ASYNCcnt path, no VGPR bounce).
// Each of the 32 lanes carries 16B per instruction; 32 issues move 16KB.
// ---------------------------------------------------------------------------
__device__ __forceinline__ void stage_tap_async(const float* __restrict__ Wt,
                                                int tap, float* bufGeneric,
                                                int lane) {
    __attribute__((address_space(3))) float* p3 =
        (__attribute__((address_space(3))) float*)bufGeneric;
    unsigned lds0 = (unsigned)(size_t)p3 + (unsigned)(lane * 16);
    const float* g = Wt + (size_t)tap * 4096 + lane * 4;
#pragma unroll
    for (int i = 0; i < 32; ++i) {
        unsigned loff = lds0 + (unsigned)(i * 512);
        const float* gp = g + i * 128;
        asm volatile("global_load_async_to_lds_b128 %0, %1, off"
                     :: "v"(loff), "v"(gp) : "memory");
    }
}

__device__ __forceinline__ void wait_async0() {
    asm volatile("s_wait_asynccnt 0x0" ::: "memory");
}

// ---------------------------------------------------------------------------
// Stage 4: WMMA GEMM.  One wave handles a 16-row tile x all 64 output cols.
// For each of 9 taps:  y = A_tile @ W_tap  via chained v_wmma_f32_16x16x4_f32
// (K=64 -> 16 steps), then complex-phase epilogue per row:
//   accR += eR*yR - eI*yI ;  accI += eR*yI + eI*yR
// W_tap matrices are double-buffered in LDS via async-to-LDS DMA so tap t+1's
// transfer overlaps tap t's WMMA chain.
// C layout: vgpr v, lane l: row M = v + 8*(l>=16), col N = l%16 (+16*nt).
// ---------------------------------------------------------------------------
__global__ void __launch_bounds__(32) k_gemm(const float* __restrict__ Are,
                                             const float* __restrict__ Aim,
                                             const float* __restrict__ Wt,
                                             float* __restrict__ OutRe,
                                             float* __restrict__ OutIm) {
    __shared__ float Bt[2][4096];    // double-buffered W_tap, [n][k] layout
    int lane = threadIdx.x;
    int r  = lane & 15;
    int hi = lane >> 4;
    int rowBase  = blockIdx.x * 16;
    int flinBase = rowBase % FREQ;   // tiles never cross batch (FREQ % 16 == 0)

    // Load A fragments once: lane<16 holds K = 4k..4k+1, lane>=16 holds 4k+2..4k+3
    vf2 aRe[16], aIm[16];
    const float* arp = Are + (size_t)(rowBase + r) * 64 + 2 * hi;
    const float* aip = Aim + (size_t)(rowBase + r) * 64 + 2 * hi;
#pragma unroll
    for (int k = 0; k < 16; ++k) {
        aRe[k] = *(const vf2*)(arp + 4 * k);
        aIm[k] = *(const vf2*)(aip + 4 * k);
    }

    vf8 accR[4], accI[4];
#pragma unroll
    for (int nt = 0; nt < 4; ++nt) { accR[nt] = 0.0f; accI[nt] = 0.0f; }

    // prologue: DMA tap 0 into buffer 0
    stage_tap_async(Wt, 0, &Bt[0][0], lane);
    wait_async0();
    __syncthreads();

    for (int tap = 0; tap < 9; ++tap) {
        int buf = tap & 1;
        // overlap: DMA next tap into the other buffer while we compute
        if (tap < 8) stage_tap_async(Wt, tap + 1, &Bt[buf ^ 1][0], lane);

        int ky = tap / 3, kx = tap % 3;
        // per-row phases of kernel-FFT tap at (shifted-index) frequency
        float eR[8], eI[8];
#pragma unroll
        for (int v = 0; v < 8; ++v) {
            int flin = flinBase + v + 8 * hi;
            int fy = flin / WF, fx = flin % WF;
            float ang = -(PI2 / 128.0f) * (float)(fy * ky + fx * kx);
            __sincosf(ang, &eI[v], &eR[v]);
        }

#pragma unroll
        for (int nt = 0; nt < 4; ++nt) {
            vf8 yR = 0.0f, yI = 0.0f;
            const float* bp = &Bt[buf][0] + (size_t)(nt * 16 + r) * 64 + 2 * hi;
#pragma unroll
            for (int k = 0; k < 16; ++k) {
                vf2 bf = *(const vf2*)(bp + 4 * k);   // ds_load_b64
                yR = __builtin_amdgcn_wmma_f32_16x16x4_f32(
                        false, aRe[k], false, bf, (short)0, yR, false, false);
                yI = __builtin_amdgcn_wmma_f32_16x16x4_f32(
                        false, aIm[k], false, bf, (short)0, yI, false, false);
            }
#pragma unroll
            for (int v = 0; v < 8; ++v) {
                accR[nt][v] += eR[v] * yR[v] - eI[v] * yI[v];
                accI[nt][v] += eR[v] * yI[v] + eI[v] * yR[v];
            }
        }

        // next buffer must be fully landed in LDS before next iteration reads it
        wait_async0();
        __syncthreads();
    }

    // store
#pragma unroll
    for (int nt = 0; nt < 4; ++nt) {
#pragma unroll
        for (int v = 0; v < 8; ++v) {
            size_t row = (size_t)rowBase + v + 8 * hi;
            size_t o   = (size_t)(nt * 16 + r);
            OutRe[row * 64 + o] = accR[nt][v];
            OutIm[row * 64 + o] = accI[nt][v];
        }
    }
}

// ---------------------------------------------------------------------------
// Stage 5: inverse column FFT (over fy), scale 1/128.
// Ycol[(bo*128+row)*65+fx]
// ---------------------------------------------------------------------------
__global__ void __launch_bounds__(64) k_ifft_cols(const float* __restrict__ OutRe,
                                                  const float* __restrict__ OutIm,
                                                  float2* __restrict__ Ycol) {
    __shared__ float2 s[128];
    int blk = blockIdx.x;            // bo*65 + fx
    int bo = blk / WF, fx = blk % WF;
    int b = bo >> 6, o = bo & 63;
    int t = threadIdx.x;
#pragma unroll
    for (int i = 0; i < 2; ++i) {
        int fy = t + (i << 6);
        size_t row = (size_t)b * FREQ + (size_t)fy * WF + fx;
        s[fy] = make_float2(OutRe[row * 64 + o], OutIm[row * 64 + o]);
    }
    fft128_shared(s, t, 1.0f);
    float2* out = Ycol + (size_t)bo * FREQ + fx;
#pragma unroll
    for (int i = 0; i < 2; ++i) {
        int rr = t + (i << 6);
        float2 v = s[rr];
        out[(size_t)rr * WF] = make_float2(v.x * (1.0f / 128.0f),
                                           v.y * (1.0f / 128.0f));
    }
}

// ---------------------------------------------------------------------------
// Stage 6: inverse row rFFT via Hermitian expansion, + bias, write output.
// ---------------------------------------------------------------------------
__global__ void __launch_bounds__(64) k_irfft_rows(const float2* __restrict__ Ycol,
                                                   const float* __restrict__ bias,
                                                   float* __restrict__ out) {
    __shared__ float2 s[128];
    int blk = blockIdx.x;            // bo*128 + row
    int bo = blk >> 7;
    int o = bo & 63;
    int t = threadIdx.x;
    const float2* src = Ycol + (size_t)bo * FREQ + (size_t)(blk & 127) * WF;
#pragma unroll
    for (int i = 0; i < 2; ++i) {
        int k = t + (i << 6);
        if (k <= 64) {
            s[k] = src[k];
        } else {
            float2 v = src[128 - k];
            s[k] = make_float2(v.x, -v.y);
        }
    }
    fft128_shared(s, t, 1.0f);
    float bv = bias[o];
    float* dst = out + (size_t)blk * WW;
#pragma unroll
    for (int i = 0; i < 2; ++i) {
        int c = t + (i << 6);
        dst[c] = s[c].x * (1.0f / 128.0f) + bv;
    }
}

// ---------------------------------------------------------------------------
// Host launcher.  Workspace layout (floats), buffers reused across stages:
//   [0, 2129920)            A_re     -> later reused by Ycol (float2)
//   [2129920, 4259840)      A_im        (Ycol spans both A halves)
//   [4259840, 8519680)      Xrow (float2) -> later reused by OutRe/OutIm
//   [8519680, 8556544)      Wt (9*64*64)
// total ~34.2 MB
// ---------------------------------------------------------------------------
extern "C" void kernel_launch(void* const* d_in, const int* in_sizes, int n_in,
                              void* d_out, int out_size, void* d_ws, size_t ws_size,
                              hipStream_t stream) {
    const float* x      = (const float*)d_in[0];
    const float* weight = (const float*)d_in[1];
    const float* bias   = (const float*)d_in[2];
    float* out = (float*)d_out;
    float* ws  = (float*)d_ws;

    const size_t N_A   = (size_t)MROWS * 64;            // 2,129,920 floats
    float*  Are   = ws;
    float*  Aim   = ws + N_A;
    float2* Xrow  = (float2*)(ws + 2 * N_A);            // 2,129,920 float2
    float*  Wt    = ws + 4 * N_A;
    float*  OutRe = ws + 2 * N_A;                       // reuse Xrow region
    float*  OutIm = ws + 3 * N_A;
    float2* Ycol  = (float2*)ws;                        // reuse A region

    // 1. transpose weights per tap
    k_prep_w<<<(9 * 64 * 64 + 255) / 256, 256, 0, stream>>>(weight, Wt);
    // 2. row rFFT of x
    k_fft_rows<<<BB * CIN * HH, 64, 0, stream>>>(x, Xrow);
    // 3. column FFT + shift + mask -> A matrices
    k_fft_cols<<<BB * CIN * WF, 64, 0, stream>>>(Xrow, Are, Aim);
    // 4. WMMA channel GEMM with 9-tap phase epilogue, async-LDS double buffer
    k_gemm<<<MROWS / 16, 32, 0, stream>>>(Are, Aim, Wt, OutRe, OutIm);
    // 5. inverse column FFT
    k_ifft_cols<<<BB * COUT * WF, 64, 0, stream>>>(OutRe, OutIm, Ycol);
    // 6. inverse row rFFT + bias
    k_irfft_rows<<<BB * COUT * HH, 64, 0, stream>>>(Ycol, bias, out);
}